// MDNRNNReward_13460427505724
// MI455X (gfx1250) — compile-verified
//
#include <hip/hip_runtime.h>
#include <hip/hip_bf16.h>
#include <math.h>

// Problem constants (match reference)
#define NROWS 16384
#define KMIX  5
#define DDIM  1088

typedef __attribute__((ext_vector_type(2))) float v2f;
typedef __attribute__((ext_vector_type(8))) float v8f;

__global__ void zero_out_kernel(float* out) { out[0] = 0.0f; }

// One wave (32 lanes) per row n. 256 threads = 8 waves = 8 rows per block.
__global__ __launch_bounds__(256)
void mdn_loss_kernel(const float* __restrict__ target,
                     const float* __restrict__ s_mean,
                     const float* __restrict__ s_logstd,
                     const float* __restrict__ lmc,
                     float* __restrict__ out) {
    // per-wave scratch: [wave][lane][slot] ; slots 5..15 zeroed so the
    // WMMA B-tile columns >= KMIX contribute nothing.
    __shared__ float red[8][32][16];

    const int lane = threadIdx.x & 31;
    const int wave = threadIdx.x >> 5;
    const int n    = blockIdx.x * 8 + wave;   // row index, < NROWS

    const v2f* tgt2 = (const v2f*)(target)   + (size_t)n * (DDIM / 2);
    const v2f* mn2  = (const v2f*)(s_mean)   + (size_t)n * (KMIX * DDIM / 2);
    const v2f* ls2  = (const v2f*)(s_logstd) + (size_t)n * (KMIX * DDIM / 2);

    float acc[KMIX];
#pragma unroll
    for (int k = 0; k < KMIX; ++k) acc[k] = 0.0f;

    // Coalesced streaming: each lane reads float2, wave covers 64 floats/iter.
    // 1088/64 = 17 iterations exactly. target is loaded once per chunk and
    // reused across the 5 components -> total traffic = 11 * N * D * 4 bytes,
    // the roofline minimum.
#pragma unroll 1
    for (int i = 0; i < DDIM / 64; ++i) {
        const int j = lane + 32 * i;
        const v2f t = tgt2[j];
#pragma unroll
        for (int k = 0; k < KMIX; ++k) {
            const v2f m = mn2[k * (DDIM / 2) + j];
            const v2f l = ls2[k * (DDIM / 2) + j];
            const float dx = t.x - m.x;
            const float dy = t.y - m.y;
            const float ex = __expf(-2.0f * l.x);   // 1/sigma^2
            const float ey = __expf(-2.0f * l.y);
            // acc += 0.5*diff^2/var + logstd   (negated later)
            acc[k] += fmaf(0.5f * dx * dx, ex, l.x)
                    + fmaf(0.5f * dy * dy, ey, l.y);
        }
    }

    // Publish per-lane partials (pad slots are zero).
#pragma unroll
    for (int k = 0; k < KMIX; ++k) red[wave][lane][k] = acc[k];
#pragma unroll
    for (int k = KMIX; k < 16; ++k) red[wave][lane][k] = 0.0f;
    __syncthreads();

    // Cross-lane reduction via V_WMMA_F32_16X16X4_F32 (f32 -> exact):
    //   D = ones(16x4) * B(4x16) + C ; chained over 8 chunks of 4 source
    // lanes, so D[m][k] = sum over all 32 lanes of acc[k].
    const int hi  = lane >> 4;     // which half of the wave
    const int col = lane & 15;     // B/D column (component index k)
    v2f aones; aones.x = 1.0f; aones.y = 1.0f;   // A = all ones
    v8f dmat = {};
#pragma unroll
    for (int c = 0; c < 8; ++c) {
        const int src = 4 * c + 2 * hi;          // B rows {0|2}, {1|3} layout
        v2f b;
        b.x = red[wave][src][col];
        b.y = red[wave][src + 1][col];
        dmat = __builtin_amdgcn_wmma_f32_16x16x4_f32(
            /*neg_a=*/false, aones, /*neg_b=*/false, b,
            /*c_mod=*/(short)0, dmat, /*reuse_a=*/false, /*reuse_b=*/false);
    }
    // Every row of D is identical; lane L holds S[L&15] in dmat[0].
    const float S = dmat[0];

    // logsumexp over the 5 components: lanes 0..4 are valid, rest -inf.
    float tval;
    {
        const int safe = (lane < KMIX) ? lane : 0;
        const float c0 = lmc[(size_t)n * KMIX + safe];
        tval = (lane < KMIX) ? (c0 - S) : -INFINITY;
    }
    float mx = tval;
#pragma unroll
    for (int off = 16; off >= 1; off >>= 1)
        mx = fmaxf(mx, __shfl_xor(mx, off, 32));
    float e = __expf(tval - mx);                  // exp(-inf) = 0 for pads
#pragma unroll
    for (int off = 16; off >= 1; off >>= 1)
        e += __shfl_xor(e, off, 32);

    if (lane == 0) {
        const float lse = mx + __logf(e);
        atomicAdd(out, -lse * (1.0f / (float)NROWS));
    }
}

extern "C" void kernel_launch(void* const* d_in, const int* in_sizes, int n_in,
                              void* d_out, int out_size, void* d_ws, size_t ws_size,
                              hipStream_t stream) {
    (void)in_sizes; (void)n_in; (void)out_size; (void)d_ws; (void)ws_size;
    const float* target   = (const float*)d_in[0];
    const float* s_mean   = (const float*)d_in[1];
    const float* s_logstd = (const float*)d_in[2];
    const float* lmc      = (const float*)d_in[3];
    float* out = (float*)d_out;

    zero_out_kernel<<<1, 1, 0, stream>>>(out);
    mdn_loss_kernel<<<dim3(NROWS / 8), dim3(256), 0, stream>>>(
        target, s_mean, s_logstd, lmc, out);
}